// PagedAttention_13838384627866
// MI455X (gfx1250) — compile-verified
//
#include <hip/hip_runtime.h>
#include <math.h>

typedef __attribute__((ext_vector_type(2))) float v2f;
typedef __attribute__((ext_vector_type(8))) float v8f;

#define BATCH     32
#define NUM_HEADS 32
#define KV_HEADS  8
#define GQA       4
#define HEAD      128
#define BS        16
#define BPS       128
#define NWAVE     4
#define LDK       132                 // padded LDS row stride (floats) -> conflict-free b64 reads
#define SCALE     0.08838834764831845f

// ---------------------------------------------------------------------------
// Stage 1: RoPE for q and k_new -> workspace (cannot mutate inputs in place)
// ---------------------------------------------------------------------------
__global__ void __launch_bounds__(HEAD)
rope_kernel(const float* __restrict__ q, const float* __restrict__ k,
            const int* __restrict__ ctx, float* __restrict__ qo,
            float* __restrict__ ko) {
  const int blk = blockIdx.x;
  const int b = blk / (NUM_HEADS + KV_HEADS);
  const int h = blk % (NUM_HEADS + KV_HEADS);
  const int d = threadIdx.x;
  const float pos = (float)(ctx[b] + 1);
  const int j = d & 63;
  const float inv = expf(-(float)j * 0.14391156605212956f); // ln(1e4)/64
  const float ang = pos * inv;
  const float cs = cosf(ang);
  const float sn = sinf(ang);
  const float* x;
  float* o;
  if (h < NUM_HEADS) {
    x = q + ((size_t)b * NUM_HEADS + h) * HEAD;
    o = qo + ((size_t)b * NUM_HEADS + h) * HEAD;
  } else {
    const int kh = h - NUM_HEADS;
    x = k + ((size_t)b * KV_HEADS + kh) * HEAD;
    o = ko + ((size_t)b * KV_HEADS + kh) * HEAD;
  }
  const float xv = x[d];
  const float rot = (d < 64) ? -x[d + 64] : x[d - 64];
  o[d] = xv * cs + rot * sn;
}

// ---------------------------------------------------------------------------
// CDNA5 async copy helpers (ASYNCcnt-tracked global -> LDS DMA path)
// ---------------------------------------------------------------------------
__device__ __forceinline__ void async_b128(unsigned lds, unsigned long long ga) {
  asm volatile("global_load_async_to_lds_b128 %0, %1, off"
               :: "v"(lds), "v"(ga) : "memory");
}
#define WAIT_ASYNC(n) asm volatile("s_wait_asynccnt %0" :: "i"(n) : "memory")

// stage one 16-key cache block (16 rows x 512B) of K or V into a padded LDS tile
__device__ __forceinline__ void stage_tile(const float* __restrict__ cache,
                                           const float* buf, int blkid, int kvh,
                                           int lane) {
  unsigned long long g = (unsigned long long)(const void*)cache
      + ((unsigned long long)blkid * (BS * KV_HEADS * HEAD)
         + (unsigned long long)(kvh * HEAD)) * 4ull
      + (unsigned)(lane * 16);
  unsigned l = (unsigned)(unsigned long long)(const void*)buf + (unsigned)(lane * 16);
#pragma unroll
  for (int r = 0; r < BS; ++r)
    async_b128(l + r * (LDK * 4), g + (unsigned long long)r * (KV_HEADS * HEAD * 4));
}

// ---------------------------------------------------------------------------
// Stage 2: flash-decoding paged attention, 1 WG per (batch, kv_head), 4 waves
// ---------------------------------------------------------------------------
__global__ void __launch_bounds__(NWAVE * 32)
attn_kernel(const float* __restrict__ v_new, const float* __restrict__ k_cache,
            const float* __restrict__ v_cache, const int* __restrict__ btab,
            const int* __restrict__ ctxl, const float* __restrict__ qro,
            const float* __restrict__ kro, float* __restrict__ out) {
  __shared__ __align__(16) float qs[16 * LDK];               // Q^T tile, rows>=G zeroed
  __shared__ __align__(16) float kt[NWAVE][2][BS * LDK];     // double-buffered K tiles
  __shared__ __align__(16) float vt[NWAVE][2][BS * LDK];     // double-buffered V tiles
  __shared__ __align__(16) float om[NWAVE][GQA][HEAD];       // per-wave partial O
  __shared__ float msh[NWAVE][GQA];
  __shared__ float lsh[NWAVE][GQA];

  const int b    = blockIdx.x / KV_HEADS;
  const int kvh  = blockIdx.x % KV_HEADS;
  const int tid  = threadIdx.x;
  const int lane = tid & 31;
  const int w    = tid >> 5;
  const int rsel = lane & 15;
  const int hsel = lane >> 4;

  const int ctx   = ctxl[b];
  const int nkeys = ctx + 1;
  const int nt    = (nkeys + BS - 1) / BS;

  // zero Q rows G..15 and fill rows 0..G-1 with roped q (disjoint -> one barrier)
  for (int i = GQA * LDK + tid; i < 16 * LDK; i += NWAVE * 32) qs[i] = 0.0f;
  {
    const int g  = tid >> 5;
    const int d4 = (tid & 31) * 4;
    const float* src = qro + (((size_t)b * NUM_HEADS) + kvh * GQA + g) * HEAD + d4;
#pragma unroll
    for (int i = 0; i < 4; ++i) qs[g * LDK + d4 + i] = src[i];
  }
  __syncthreads();

  const v8f vzero = {0.f, 0.f, 0.f, 0.f, 0.f, 0.f, 0.f, 0.f};
  v8f oacc[8];
#pragma unroll
  for (int dt = 0; dt < 8; ++dt) oacc[dt] = vzero;
  float mrun = -__builtin_inff();
  float lrun = 0.0f;

  // prefetch first two tiles for this wave
  if (w < nt) {
    const int blk0 = btab[b * BPS + w];
    stage_tile(k_cache, &kt[w][0][0], blk0, kvh, lane);
    stage_tile(v_cache, &vt[w][0][0], blk0, kvh, lane);
  }
  if (w + NWAVE < nt) {
    const int blk1 = btab[b * BPS + w + NWAVE];
    stage_tile(k_cache, &kt[w][1][0], blk1, kvh, lane);
    stage_tile(v_cache, &vt[w][1][0], blk1, kvh, lane);
  }

  int bufsel = 0;
  for (int it = w; it < nt; it += NWAVE) {
    if (it + NWAVE < nt) { WAIT_ASYNC(32); } else { WAIT_ASYNC(0); }
    float* K = &kt[w][bufsel][0];
    float* V = &vt[w][bufsel][0];
    const int t0 = it * BS;

    // patch the freshly-written token row (t == ctx) with roped k / new v
    if (ctx >= t0 && ctx < t0 + BS) {
      const int r = ctx - t0;
      const float* krow = kro   + ((size_t)b * KV_HEADS + kvh) * HEAD + lane * 4;
      const float* vrow = v_new + ((size_t)b * KV_HEADS + kvh) * HEAD + lane * 4;
#pragma unroll
      for (int i = 0; i < 4; ++i) {
        K[r * LDK + lane * 4 + i] = krow[i];
        V[r * LDK + lane * 4 + i] = vrow[i];
      }
    }

    // ---- S^T(16t x 16g) = K_tile(16x4) x Q^T(4x16), 32 chunked f32 WMMAs ----
    v8f s = vzero;
#pragma unroll
    for (int c = 0; c < 32; ++c) {
      const v2f a  = *(const v2f*)&K [rsel * LDK + 4 * c + 2 * hsel];
      const v2f bq = *(const v2f*)&qs[rsel * LDK + 4 * c + 2 * hsel];
      s = __builtin_amdgcn_wmma_f32_16x16x4_f32(false, a, false, bq, (short)0, s,
                                                false, false);
    }

    // scale + causal/length mask (row t = t0 + j + 8*hsel)
#pragma unroll
    for (int j = 0; j < 8; ++j) {
      const int tg = t0 + j + 8 * hsel;
      const float sv = s[j] * SCALE;
      s[j] = (tg < nkeys) ? sv : -__builtin_inff();
    }

    // online softmax per column g (= rsel); halves combined via shfl_xor 16
    float mt = s[0];
#pragma unroll
    for (int j = 1; j < 8; ++j) mt = fmaxf(mt, s[j]);
    mt = fmaxf(mt, __shfl_xor(mt, 16));
    const float mnew  = fmaxf(mrun, mt);
    const float alpha = __expf(mrun - mnew);
    float p[8];
    float lt = 0.0f;
#pragma unroll
    for (int j = 0; j < 8; ++j) { p[j] = __expf(s[j] - mnew); lt += p[j]; }
    lt += __shfl_xor(lt, 16);
    lrun = lrun * alpha + lt;
    mrun = mnew;
#pragma unroll
    for (int dt = 0; dt < 8; ++dt) oacc[dt] *= alpha;

    // ---- O^T(16d x 16g) += V^T(16x4) x P(4x16), 32 f32 WMMAs ----
    // re-layout P (C-layout) into B-operand layout with half-wave swaps
    float xp[8];
#pragma unroll
    for (int j = 0; j < 8; ++j) xp[j] = __shfl_xor(p[j], 16);
    const bool hi = (hsel != 0);
    v2f bb[4];
    bb[0].x = hi ? xp[2] : p[0];  bb[0].y = hi ? xp[3] : p[1];   // t = 0..3
    bb[1].x = hi ? xp[6] : p[4];  bb[1].y = hi ? xp[7] : p[5];   // t = 4..7
    bb[2].x = hi ? p[2] : xp[0];  bb[2].y = hi ? p[3] : xp[1];   // t = 8..11
    bb[3].x = hi ? p[6] : xp[4];  bb[3].y = hi ? p[7] : xp[5];   // t = 12..15
#pragma unroll
    for (int tc = 0; tc < 4; ++tc) {
      const int trow = 4 * tc + 2 * hsel;
#pragma unroll
      for (int dt = 0; dt < 8; ++dt) {
        v2f a;
        a.x = V[ trow      * LDK + dt * 16 + rsel];
        a.y = V[(trow + 1) * LDK + dt * 16 + rsel];
        oacc[dt] = __builtin_amdgcn_wmma_f32_16x16x4_f32(false, a, false, bb[tc],
                                                         (short)0, oacc[dt],
                                                         false, false);
      }
    }

    // prefetch tile it+8 into the buffer just consumed
    if (it + 2 * NWAVE < nt) {
      const int nb = btab[b * BPS + it + 2 * NWAVE];
      stage_tile(k_cache, &kt[w][bufsel][0], nb, kvh, lane);
      stage_tile(v_cache, &vt[w][bufsel][0], nb, kvh, lane);
    }
    bufsel ^= 1;
  }

  // dump per-wave partial (m, l, O^T)
  if (rsel < GQA) {
#pragma unroll
    for (int dt = 0; dt < 8; ++dt)
#pragma unroll
      for (int j = 0; j < 8; ++j)
        om[w][rsel][dt * 16 + j + 8 * hsel] = oacc[dt][j];
    if (hsel == 0) { msh[w][rsel] = mrun; lsh[w][rsel] = lrun; }
  }
  __syncthreads();

  // cross-wave combine and write out[b][kvh*G+g][d]
#pragma unroll
  for (int rep = 0; rep < 4; ++rep) {
    const int lin = tid + rep * 128;
    const int g = lin >> 7;
    const int d = lin & 127;
    float M = msh[0][g];
#pragma unroll
    for (int wv = 1; wv < NWAVE; ++wv) M = fmaxf(M, msh[wv][g]);
    float num = 0.0f, den = 0.0f;
#pragma unroll
    for (int wv = 0; wv < NWAVE; ++wv) {
      const float f = __expf(msh[wv][g] - M);
      den += f * lsh[wv][g];
      num += f * om[wv][g][d];
    }
    out[(((size_t)b * NUM_HEADS) + kvh * GQA + g) * HEAD + d] = num / den;
  }
}

// ---------------------------------------------------------------------------
extern "C" void kernel_launch(void* const* d_in, const int* in_sizes, int n_in,
                              void* d_out, int out_size, void* d_ws, size_t ws_size,
                              hipStream_t stream) {
  (void)in_sizes; (void)n_in; (void)out_size; (void)ws_size;
  const float* q  = (const float*)d_in[0];
  const float* k  = (const float*)d_in[1];
  const float* v  = (const float*)d_in[2];
  const float* kc = (const float*)d_in[3];
  const float* vc = (const float*)d_in[4];
  const int* bt   = (const int*)d_in[5];
  const int* cl   = (const int*)d_in[6];
  float* out = (float*)d_out;

  float* qro = (float*)d_ws;                       // B*32*128 roped q
  float* kro = qro + (size_t)BATCH * NUM_HEADS * HEAD; // B*8*128 roped k

  rope_kernel<<<BATCH * (NUM_HEADS + KV_HEADS), HEAD, 0, stream>>>(q, k, cl, qro, kro);
  attn_kernel<<<BATCH * KV_HEADS, NWAVE * 32, 0, stream>>>(v, kc, vc, bt, cl,
                                                           qro, kro, out);
}